// GraphSAGE_75479755259985
// MI455X (gfx1250) — compile-verified
//
#include <hip/hip_runtime.h>

typedef __attribute__((ext_vector_type(2))) float v2f;
typedef __attribute__((ext_vector_type(8))) float v8f;

#define N_NODES   40000
#define N_EDGES   640000
#define IN_FEATS  128
#define H_FEATS   256
#define OUT_FEATS 10
#define N_GRAPHS  64

// ---------------- degree / graph counts ----------------
__global__ void count_edges_kernel(const int* __restrict__ dst, float* __restrict__ cnt, int nE) {
    int e = blockIdx.x * blockDim.x + threadIdx.x;
    if (e < nE) atomicAdd(&cnt[dst[e]], 1.0f);
}

__global__ void inv_count_kernel(const float* __restrict__ cnt, float* __restrict__ invc, int n) {
    int i = blockIdx.x * blockDim.x + threadIdx.x;
    if (i < n) invc[i] = 1.0f / fmaxf(cnt[i], 1.0f);
}

__global__ void graph_count_kernel(const int* __restrict__ batch, float* __restrict__ gcnt, int n) {
    int i = blockIdx.x * blockDim.x + threadIdx.x;
    if (i < n) atomicAdd(&gcnt[batch[i]], 1.0f);
}

// ---------------- weight repack: K-pair interleave ----------------
// Wp[kk*256 + n] (float2) = { W[2kk][n], W[2kk+1][n] }  -> one global_load_b64 per B fragment
__global__ void repack_w_kernel(const float* __restrict__ W, float* __restrict__ Wp, int K) {
    int t = blockIdx.x * blockDim.x + threadIdx.x;
    int total = (K >> 1) * H_FEATS;
    if (t >= total) return;
    int kk = t >> 8;          // / 256
    int n  = t & 255;
    float2 v;
    v.x = W[(2 * kk) * H_FEATS + n];
    v.y = W[(2 * kk + 1) * H_FEATS + n];
    reinterpret_cast<float2*>(Wp)[t] = v;
}

// ---------------- edge scatter-add: one wave32 per edge ----------------
// Lane l moves float4 chunk(s) of the source row onto the dest row. Edge indices are
// loaded once per wave (uniform address -> single memory request).
__global__ __launch_bounds__(256) void scatter_add_kernel(
    const float* __restrict__ x, const int* __restrict__ src,
    const int* __restrict__ dst, float* __restrict__ agg, int nE, int K) {
    const int lane = threadIdx.x & 31;
    const int wave = threadIdx.x >> 5;
    const int e = blockIdx.x * 8 + wave;
    if (e >= nE) return;
    const int s = src[e], d = dst[e];
    const float4* xr = reinterpret_cast<const float4*>(x + (long long)s * K);
    float* o = agg + (long long)d * K;
    const int c4 = K >> 2;
    for (int c = lane; c < c4; c += 32) {
        const float4 v = xr[c];
        float* p = o + 4 * c;
        atomicAdd(p + 0, v.x);
        atomicAdd(p + 1, v.y);
        atomicAdd(p + 2, v.z);
        atomicAdd(p + 3, v.w);
    }
}

// ---------------- fused SAGE layer ----------------
// out = l2norm_rows((agg*invc)@Wl + b + x@Wr) -> ReLU -> store (or fused mean-pool atomics)
// Block = 256 threads (8 wave32), owns 32 rows x 256 cols.
//  * ONE 32 x (K+4) LDS buffer, reused: stage agg*invc -> GEMM1 -> barrier -> stage x -> GEMM2.
//    Pitch K+4 floats -> conflict-free ds_load_2addr A-fragment reads.
//  * Wave w owns cols [32w,32w+32): 2x2 grid of 16x16 v_wmma_f32_16x16x4_f32 tiles.
//  * B fragments come from K-pair repacked weights: one b64 load per fragment.
// Fragment layouts per ISA 7.12.2:
//  A 16x4 : lanes 0-15 -> M=lane, K={k,k+1}; lanes 16-31 -> M=lane-16, K={k+2,k+3}
//  B 4x16 : lanes 0-15 -> N=lane, K={k,k+1}; lanes 16-31 -> N=lane-16, K={k+2,k+3}
//  C 16x16: vgpr i, lanes 0-15 -> row i; lanes 16-31 -> row i+8; col = lane&15
__global__ __launch_bounds__(256) void sage_layer_kernel(
    const float* __restrict__ agg, const float* __restrict__ invc,
    const float* __restrict__ xin,
    const float* __restrict__ Wpl, const float* __restrict__ Wpr,   // repacked [K/2][256] float2
    const float* __restrict__ bias,
    float* __restrict__ out,            // [M,256] (used when pooled == nullptr)
    const int* __restrict__ batch,      // [M]     (used when pooled != nullptr)
    float* __restrict__ pooled,         // [64,256] accumulate, or nullptr
    int K, int cshift)                  // cshift = log2(K/4)
{
    const int N = H_FEATS;              // 256
    const int P = K + 4;                // LDS pitch (floats)
    extern __shared__ float smem[];
    float* sA      = smem;              // 32 * P (staged operand, reused)
    float* s_sumsq = sA + 32 * P;       // 32

    const int m0    = blockIdx.x * 32;
    const int tid   = threadIdx.x;
    const int lane  = tid & 31;
    const int wave  = tid >> 5;         // 0..7
    const int nidx  = lane & 15;
    const int khalf = (lane >> 4) << 1; // 0 or 2
    const int rbase = (lane >> 4) << 3; // 0 or 8
    const int nb    = wave * 32;
    const int c4    = K >> 2;

    if (tid < 32) s_sumsq[tid] = 0.0f;

    // ---- stage 1: scaled neighbor aggregate ----
    for (int f = tid; f < 32 * c4; f += 256) {
        const int row = f >> cshift;
        const int cc  = f & (c4 - 1);
        const float sc = invc[m0 + row];
        const float4 va = reinterpret_cast<const float4*>(agg + (long long)(m0 + row) * K)[cc];
        float* da = sA + row * P + 4 * cc;
        da[0] = va.x * sc; da[1] = va.y * sc; da[2] = va.z * sc; da[3] = va.w * sc;
    }
    __syncthreads();

    v8f c00 = {}, c01 = {}, c10 = {}, c11 = {};
    const v2f* wpl = reinterpret_cast<const v2f*>(Wpl);
    const v2f* wpr = reinterpret_cast<const v2f*>(Wpr);

    // ---- GEMM 1: (agg*invc) @ Wl ----
    for (int k = 0; k < K; k += 4) {
        const int kb = k + khalf;
        const int kk = (k >> 1) + (lane >> 4);
        v2f a0; a0.x = sA[nidx * P + kb];        a0.y = sA[nidx * P + kb + 1];
        v2f a1; a1.x = sA[(16 + nidx) * P + kb]; a1.y = sA[(16 + nidx) * P + kb + 1];
        const v2f* wrow = wpl + kk * N + nb + nidx;
        const v2f b0 = wrow[0];
        const v2f b1 = wrow[16];
        c00 = __builtin_amdgcn_wmma_f32_16x16x4_f32(false, a0, false, b0, (short)0, c00, false, false);
        c01 = __builtin_amdgcn_wmma_f32_16x16x4_f32(false, a0, false, b1, (short)0, c01, false, false);
        c10 = __builtin_amdgcn_wmma_f32_16x16x4_f32(false, a1, false, b0, (short)0, c10, false, false);
        c11 = __builtin_amdgcn_wmma_f32_16x16x4_f32(false, a1, false, b1, (short)0, c11, false, false);
    }
    __syncthreads();

    // ---- stage 2: root features (reuse same LDS buffer) ----
    for (int f = tid; f < 32 * c4; f += 256) {
        const int row = f >> cshift;
        const int cc  = f & (c4 - 1);
        const float4 vx = reinterpret_cast<const float4*>(xin + (long long)(m0 + row) * K)[cc];
        float* dx = sA + row * P + 4 * cc;
        dx[0] = vx.x; dx[1] = vx.y; dx[2] = vx.z; dx[3] = vx.w;
    }
    __syncthreads();

    // ---- GEMM 2: x @ Wr (accumulate) ----
    for (int k = 0; k < K; k += 4) {
        const int kb = k + khalf;
        const int kk = (k >> 1) + (lane >> 4);
        v2f a0; a0.x = sA[nidx * P + kb];        a0.y = sA[nidx * P + kb + 1];
        v2f a1; a1.x = sA[(16 + nidx) * P + kb]; a1.y = sA[(16 + nidx) * P + kb + 1];
        const v2f* wrow = wpr + kk * N + nb + nidx;
        const v2f b0 = wrow[0];
        const v2f b1 = wrow[16];
        c00 = __builtin_amdgcn_wmma_f32_16x16x4_f32(false, a0, false, b0, (short)0, c00, false, false);
        c01 = __builtin_amdgcn_wmma_f32_16x16x4_f32(false, a0, false, b1, (short)0, c01, false, false);
        c10 = __builtin_amdgcn_wmma_f32_16x16x4_f32(false, a1, false, b0, (short)0, c10, false, false);
        c11 = __builtin_amdgcn_wmma_f32_16x16x4_f32(false, a1, false, b1, (short)0, c11, false, false);
    }

    // ---- epilogue: bias, row L2-norm over 256 cols (LDS fp32 atomics), ReLU, store/pool ----
    const float bc0 = bias[nb + nidx];
    const float bc1 = bias[nb + 16 + nidx];
    float v00[8], v01[8], v10[8], v11[8];
#pragma unroll
    for (int i = 0; i < 8; ++i) {
        v00[i] = c00[i] + bc0; v01[i] = c01[i] + bc1;
        v10[i] = c10[i] + bc0; v11[i] = c11[i] + bc1;
        atomicAdd(&s_sumsq[rbase + i],      v00[i] * v00[i] + v01[i] * v01[i]);
        atomicAdd(&s_sumsq[16 + rbase + i], v10[i] * v10[i] + v11[i] * v11[i]);
    }
    __syncthreads();
#pragma unroll
    for (int i = 0; i < 8; ++i) {
        const int r0 = rbase + i, r1 = 16 + rbase + i;
        const float inv0 = 1.0f / fmaxf(sqrtf(s_sumsq[r0]), 1e-12f);
        const float inv1 = 1.0f / fmaxf(sqrtf(s_sumsq[r1]), 1e-12f);
        const float a00 = fmaxf(v00[i] * inv0, 0.0f);
        const float a01 = fmaxf(v01[i] * inv0, 0.0f);
        const float a10 = fmaxf(v10[i] * inv1, 0.0f);
        const float a11 = fmaxf(v11[i] * inv1, 0.0f);
        const int gr0 = m0 + r0, gr1 = m0 + r1;
        if (pooled != nullptr) {
            const int g0 = batch[gr0], g1 = batch[gr1];
            atomicAdd(&pooled[g0 * N + nb + nidx],      a00);
            atomicAdd(&pooled[g0 * N + nb + 16 + nidx], a01);
            atomicAdd(&pooled[g1 * N + nb + nidx],      a10);
            atomicAdd(&pooled[g1 * N + nb + 16 + nidx], a11);
        } else {
            out[(long long)gr0 * N + nb + nidx]      = a00;
            out[(long long)gr0 * N + nb + 16 + nidx] = a01;
            out[(long long)gr1 * N + nb + nidx]      = a10;
            out[(long long)gr1 * N + nb + 16 + nidx] = a11;
        }
    }
}

// ---------------- final linear on pooled graph embeddings ----------------
__global__ void final_linear_kernel(const float* __restrict__ pooled, const float* __restrict__ gcnt,
                                    const float* __restrict__ W3, const float* __restrict__ b3,
                                    float* __restrict__ out) {
    int t = blockIdx.x * blockDim.x + threadIdx.x;
    if (t >= N_GRAPHS * OUT_FEATS) return;
    int g = t / OUT_FEATS, o = t % OUT_FEATS;
    float ginv = 1.0f / fmaxf(gcnt[g], 1.0f);
    const float* p = pooled + g * H_FEATS;
    float s = 0.0f;
    for (int k = 0; k < H_FEATS; ++k) s += p[k] * W3[k * OUT_FEATS + o];
    out[t] = s * ginv + b3[o];
}

extern "C" void kernel_launch(void* const* d_in, const int* in_sizes, int n_in,
                              void* d_out, int out_size, void* d_ws, size_t ws_size,
                              hipStream_t stream) {
    (void)in_sizes; (void)n_in; (void)out_size; (void)ws_size;
    const float* x    = (const float*)d_in[0];
    const int*   edge = (const int*)d_in[1];     // [2, E] row-major (int32)
    const int*   batch= (const int*)d_in[2];
    const float* W1l  = (const float*)d_in[3];
    const float* b1l  = (const float*)d_in[4];
    const float* W1r  = (const float*)d_in[5];
    const float* W2l  = (const float*)d_in[6];
    const float* b2l  = (const float*)d_in[7];
    const float* W2r  = (const float*)d_in[8];
    const float* W3   = (const float*)d_in[9];
    const float* b3   = (const float*)d_in[10];
    const int* src = edge;
    const int* dst = edge + N_EDGES;
    float* out = (float*)d_out;

    // workspace layout (fp32): cnt | invc | agg1 | h1 | agg2 | pooled | gcnt | Wp*  (~104 MB)
    float* ws     = (float*)d_ws;
    float* cnt    = ws;
    float* invc   = cnt  + N_NODES;
    float* agg1   = invc + N_NODES;
    float* h1     = agg1 + (size_t)N_NODES * IN_FEATS;
    float* agg2   = h1   + (size_t)N_NODES * H_FEATS;
    float* pooled = agg2 + (size_t)N_NODES * H_FEATS;
    float* gcnt   = pooled + (size_t)N_GRAPHS * H_FEATS;
    float* wp1l   = gcnt + N_GRAPHS;
    float* wp1r   = wp1l + (size_t)IN_FEATS * H_FEATS;
    float* wp2l   = wp1r + (size_t)IN_FEATS * H_FEATS;
    float* wp2r   = wp2l + (size_t)H_FEATS * H_FEATS;

    hipMemsetAsync(cnt,    0, (size_t)N_NODES * sizeof(float), stream);
    hipMemsetAsync(agg1,   0, (size_t)N_NODES * IN_FEATS * sizeof(float), stream);
    hipMemsetAsync(agg2,   0, (size_t)N_NODES * H_FEATS * sizeof(float), stream);
    hipMemsetAsync(pooled, 0, (size_t)(N_GRAPHS * H_FEATS + N_GRAPHS) * sizeof(float), stream);

    // weight repack (one-time per call; tiny)
    repack_w_kernel<<<(IN_FEATS / 2 * H_FEATS + 255) / 256, 256, 0, stream>>>(W1l, wp1l, IN_FEATS);
    repack_w_kernel<<<(IN_FEATS / 2 * H_FEATS + 255) / 256, 256, 0, stream>>>(W1r, wp1r, IN_FEATS);
    repack_w_kernel<<<(H_FEATS / 2 * H_FEATS + 255) / 256, 256, 0, stream>>>(W2l, wp2l, H_FEATS);
    repack_w_kernel<<<(H_FEATS / 2 * H_FEATS + 255) / 256, 256, 0, stream>>>(W2r, wp2r, H_FEATS);

    count_edges_kernel<<<(N_EDGES + 255) / 256, 256, 0, stream>>>(dst, cnt, N_EDGES);
    inv_count_kernel  <<<(N_NODES + 255) / 256, 256, 0, stream>>>(cnt, invc, N_NODES);
    graph_count_kernel<<<(N_NODES + 255) / 256, 256, 0, stream>>>(batch, gcnt, N_NODES);

    // layer 1
    scatter_add_kernel<<<(N_EDGES + 7) / 8, 256, 0, stream>>>(x, src, dst, agg1, N_EDGES, IN_FEATS);
    {
        const int P = IN_FEATS + 4;
        const size_t smem = (size_t)(32 * P + 32) * sizeof(float);   // ~17.0 KB
        sage_layer_kernel<<<N_NODES / 32, 256, smem, stream>>>(
            agg1, invc, x, wp1l, wp1r, b1l, h1, nullptr, nullptr, IN_FEATS, 5);
    }
    // layer 2 (epilogue fuses global mean-pool accumulation)
    scatter_add_kernel<<<(N_EDGES + 7) / 8, 256, 0, stream>>>(h1, src, dst, agg2, N_EDGES, H_FEATS);
    {
        const int P = H_FEATS + 4;
        const size_t smem = (size_t)(32 * P + 32) * sizeof(float);   // ~33.4 KB
        sage_layer_kernel<<<N_NODES / 32, 256, smem, stream>>>(
            agg2, invc, h1, wp2l, wp2r, b2l, nullptr, batch, pooled, H_FEATS, 6);
    }

    final_linear_kernel<<<(N_GRAPHS * OUT_FEATS + 255) / 256, 256, 0, stream>>>(
        pooled, gcnt, W3, b3, out);
}